// shapeClassifier_78348793414375
// MI455X (gfx1250) — compile-verified
//
#include <hip/hip_runtime.h>
#include <hip/hip_bf16.h>
#include <math.h>

typedef __attribute__((ext_vector_type(2))) float v2f;
typedef __attribute__((ext_vector_type(8))) float v8f;

#define EPSF 1e-7f
#define MROWS 65536
#define HID 256

__device__ __forceinline__ v8f wmma_f32_4(v2f a, v2f b, v8f c) {
  // V_WMMA_F32_16X16X4_F32 : D = A(16x4) * B(4x16) + C(16x16)
  return __builtin_amdgcn_wmma_f32_16x16x4_f32(false, a, false, b, (short)0, c,
                                               false, false);
}

__device__ __forceinline__ float sgnf(float v) {
  return (v > 0.f) ? 1.f : ((v < 0.f) ? -1.f : 0.f);
}

// ---------------------------------------------------------------------------
// Weight padding: W0 (255x256)->(256x256), W3 (256x8)->(256x16), b3 (8)->(16)
// ---------------------------------------------------------------------------
__global__ __launch_bounds__(256) void prep_kernel(
    const float* __restrict__ W0, const float* __restrict__ W3,
    const float* __restrict__ b3, float* __restrict__ W0p,
    float* __restrict__ W3p, float* __restrict__ b3p) {
  int i = blockIdx.x * blockDim.x + threadIdx.x;
  int T = gridDim.x * blockDim.x;
  for (int t = i; t < 256 * 256; t += T) {
    int r = t >> 8, c = t & 255;
    W0p[t] = (r < 255) ? W0[r * 256 + c] : 0.f;
  }
  for (int t = i; t < 256 * 16; t += T) {
    int r = t >> 4, c = t & 15;
    W3p[t] = (c < 8) ? W3[r * 8 + c] : 0.f;
  }
  if (i < 16) b3p[i] = (i < 8) ? b3[i] : 0.f;
}

// ---------------------------------------------------------------------------
// Geometry kernel: one wave32 per patch.
// Builds normalized Laplacian (21x21), Jacobi eigensolver in LDS,
// eigen sort + sign fix + row reorder + canonicalization, writes 256-wide
// padded feature row (252 features + 3 eigenvalues + 1 zero pad).
// ---------------------------------------------------------------------------
__global__ __launch_bounds__(32) void geom_kernel(const float* __restrict__ x,
                                                  float* __restrict__ feat) {
  const int m = blockIdx.x;  // patch id
  const int b = m >> 11;
  const int n = m & 2047;
  const int lane = threadIdx.x;

  __shared__ float P[21][3];
  __shared__ float A[21][22];
  __shared__ float V[21][22];
  __shared__ float colw[21];
  __shared__ float ev[21];
  __shared__ float res[21];
  __shared__ int ord[21];
  __shared__ int idxs[21];
  __shared__ float D[21][3];
  __shared__ float FIX[21][3];
  __shared__ float ROT[21][3];
  __shared__ float FIN[21][3];

  // Load points: pts[k][c] = x[b][c][n][k]
  for (int t = lane; t < 63; t += 32) {
    int k = t / 3, c = t % 3;
    P[k][c] = x[(((size_t)b * 3 + c) * 2048 + n) * 21 + k];
  }
  __syncthreads();

  // Affinity W = exp(-d2)
  for (int e = lane; e < 441; e += 32) {
    int i = e / 21, j = e % 21;
    float dx = P[i][0] - P[j][0];
    float dy = P[i][1] - P[j][1];
    float dz = P[i][2] - P[j][2];
    A[i][j] = __expf(-(dx * dx + dy * dy + dz * dz));
  }
  __syncthreads();
  if (lane < 21) {
    float s = 0.f;
    for (int i = 0; i < 21; i++) s += A[i][lane];
    colw[lane] = sqrtf(1.f / (s + EPSF));
  }
  __syncthreads();
  // L = I - inv W inv ; V = I
  for (int e = lane; e < 441; e += 32) {
    int i = e / 21, j = e % 21;
    A[i][j] = ((i == j) ? 1.f : 0.f) - colw[i] * A[i][j] * colw[j];
    V[i][j] = (i == j) ? 1.f : 0.f;
  }
  __syncthreads();

  // Cyclic Jacobi sweeps (symmetric 21x21)
  for (int sweep = 0; sweep < 9; sweep++) {
    for (int p = 0; p < 20; p++) {
      for (int q = p + 1; q < 21; q++) {
        float apq = A[p][q];
        bool doRot = fabsf(apq) > 1e-12f;
        float c = 1.f, s = 0.f;
        if (doRot) {
          float tau = (A[q][q] - A[p][p]) / (2.f * apq);
          float t = ((tau >= 0.f) ? 1.f : -1.f) /
                    (fabsf(tau) + sqrtf(1.f + tau * tau));
          c = 1.f / sqrtf(1.f + t * t);
          s = t * c;
        }
        __syncthreads();
        if (doRot && lane < 21) {  // column update: A <- A*J
          float akp = A[lane][p], akq = A[lane][q];
          A[lane][p] = c * akp - s * akq;
          A[lane][q] = s * akp + c * akq;
        }
        __syncthreads();
        if (doRot && lane < 21) {  // row update: A <- J^T*A ; V <- V*J
          float apk = A[p][lane], aqk = A[q][lane];
          A[p][lane] = c * apk - s * aqk;
          A[q][lane] = s * apk + c * aqk;
          float vkp = V[lane][p], vkq = V[lane][q];
          V[lane][p] = c * vkp - s * vkq;
          V[lane][q] = s * vkp + c * vkq;
        }
        __syncthreads();
      }
    }
  }

  // Ascending eigen sort via rank
  if (lane < 21) ev[lane] = A[lane][lane];
  __syncthreads();
  if (lane < 21) {
    int r = 0;
    float myv = ev[lane];
    for (int i = 0; i < 21; i++) {
      float vi = ev[i];
      if (vi < myv || (vi == myv && i < lane)) r++;
    }
    ord[r] = lane;
  }
  __syncthreads();
  int c1 = ord[1], c2 = ord[2], c3 = ord[3];
  float e1 = ev[c1], e2 = ev[c2], e3 = ev[c3];

  // Sign fix on first kept eigenvector
  float bestAbs = -1.f, bestVal = 0.f;
  for (int k = 0; k < 21; k++) {
    float v = V[k][c1];
    float a = fabsf(v);
    if (a > bestAbs) { bestAbs = a; bestVal = v; }
  }
  float sg = sgnf(bestVal);
  if (lane < 21) res[lane] = V[lane][c1] * sg;
  __syncthreads();

  // idx = [0, argsort(res[1:]) + 1]
  if (lane == 0) idxs[0] = 0;
  if (lane >= 1 && lane < 21) {
    int r = 0;
    float myv = res[lane];
    for (int i = 1; i < 21; i++) {
      float vi = res[i];
      if (vi < myv || (vi == myv && i < lane)) r++;
    }
    idxs[r + 1] = lane;
  }
  __syncthreads();
  if (lane < 21) {
    int src = idxs[lane];
    D[lane][0] = P[src][0];
    D[lane][1] = P[src][1];
    D[lane][2] = P[src][2];
    FIX[lane][0] = V[src][c1];
    FIX[lane][1] = V[src][c2];
    FIX[lane][2] = V[src][c3];
  }
  __syncthreads();

  // Canonicalization (Rodrigues to +z, then z-rotation)
  float mx = 0.f, my = 0.f, mz = 0.f;
  for (int k = 0; k < 21; k++) {
    mx += D[k][0]; my += D[k][1]; mz += D[k][2];
  }
  mx /= 21.f; my /= 21.f; mz /= 21.f;
  float mn = sqrtf(mx * mx + my * my + mz * mz) + EPSF;
  float v1x = mx / mn, v1y = my / mn, v1z = mz / mn;
  float axx = v1y, axy = -v1x;  // cross(v1, ez), z comp = 0
  float an = sqrtf(axx * axx + axy * axy) + EPSF;
  axx /= an; axy /= an;
  float cost = v1z;
  float sint = sqrtf(1.f - cost * cost + EPSF);
  float Kx[3][3] = {{0.f, 0.f, axy}, {0.f, 0.f, -axx}, {-axy, axx, 0.f}};
  float K2[3][3];
  for (int i2 = 0; i2 < 3; i2++)
    for (int j2 = 0; j2 < 3; j2++) {
      float s2 = 0.f;
      for (int k2 = 0; k2 < 3; k2++) s2 += Kx[i2][k2] * Kx[k2][j2];
      K2[i2][j2] = s2;
    }
  float R1[3][3];
  for (int i2 = 0; i2 < 3; i2++)
    for (int j2 = 0; j2 < 3; j2++)
      R1[i2][j2] = ((i2 == j2) ? 1.f : 0.f) + sint * Kx[i2][j2] +
                   (1.f - cost) * K2[i2][j2];

  if (lane < 21) {
    float dx = D[lane][0], dy = D[lane][1], dz = D[lane][2];
    ROT[lane][0] = dx * R1[0][0] + dy * R1[0][1] + dz * R1[0][2];
    ROT[lane][1] = dx * R1[1][0] + dy * R1[1][1] + dz * R1[1][2];
    ROT[lane][2] = dx * R1[2][0] + dy * R1[2][1] + dz * R1[2][2];
  }
  __syncthreads();
  float rpx = ROT[20][0], rpy = ROT[20][1];
  float ang = -atan2f(rpy, rpx);
  if (rpx < 0.f) ang += 3.14159265358979323846f;
  float ca = cosf(ang), sa = sinf(ang);
  if (lane < 21) {
    float rx = ROT[lane][0], ry = ROT[lane][1], rz = ROT[lane][2];
    FIN[lane][0] = ca * rx - sa * ry;
    FIN[lane][1] = sa * rx + ca * ry;
    FIN[lane][2] = rz;
  }
  __syncthreads();
  float sg2 = sgnf(FIN[1][0]);
  float frx = FIN[20][0], frz = FIN[20][2];

  float px = 0.f, py = 0.f, pz = 0.f;
  if (lane < 21) {
    if (lane == 20) { px = fabsf(frx); py = 0.f; pz = frz; }
    else {
      px = FIN[lane][0] * sg2;
      py = FIN[lane][1] * sg2;
      pz = FIN[lane][2];
    }
    if (fabsf(px) < EPSF) px = 0.f;
    if (fabsf(py) < EPSF) py = 0.f;
    if (fabsf(pz) < EPSF) pz = 0.f;
  }

  size_t base = (size_t)m * 256;
  if (lane < 21) {
    feat[base + 0 * 21 + lane] = px * px;
    feat[base + 1 * 21 + lane] = px * py;
    feat[base + 2 * 21 + lane] = px * pz;
    feat[base + 3 * 21 + lane] = py * py;
    feat[base + 4 * 21 + lane] = py * pz;
    feat[base + 5 * 21 + lane] = pz * pz;
    feat[base + 6 * 21 + lane] = px;
    feat[base + 7 * 21 + lane] = py;
    feat[base + 8 * 21 + lane] = pz;
    feat[base + 9 * 21 + lane] = FIX[lane][0];
    feat[base + 10 * 21 + lane] = FIX[lane][1];
    feat[base + 11 * 21 + lane] = FIX[lane][2];
  } else if (lane == 21) feat[base + 252] = e1;
  else if (lane == 22) feat[base + 253] = e2;
  else if (lane == 23) feat[base + 254] = e3;
  else if (lane == 24) feat[base + 255] = 0.f;
}

// ---------------------------------------------------------------------------
// Main GEMM: Y = X(MxK) * W(KxN) + bias, fp32 WMMA 16x16x4.
// Block tile 128x128, 8 waves (4 M x 2 N), wave tile 32x64 (2x4 subtiles).
// W tile stored TRANSPOSED in LDS so each B fragment is one aligned 8-byte
// contiguous LDS read (no register repacking before WMMA). Row stride 18
// floats => 8B alignment for every fragment + conflict-free bank spread.
// ---------------------------------------------------------------------------
__global__ __launch_bounds__(256) void gemm_wmma_kernel(
    const float* __restrict__ X, const float* __restrict__ Wt,
    const float* __restrict__ bias, float* __restrict__ Y, int M, int N,
    int K) {
  __shared__ float Xs[128][18];    // [row][k]  (A tile)
  __shared__ float Wst[128][18];   // [col][k]  (B tile, transposed)

  const int tid = threadIdx.x;
  const int wave = tid >> 5;
  const int lane = tid & 31;
  const int wm = wave & 3;    // wave M index (0..3)
  const int wn = wave >> 2;   // wave N index (0..1)
  const int l16 = lane & 15;
  const int half2 = (lane >> 4) << 1;  // lanes 0-15 -> k0,k1 ; 16-31 -> k2,k3
  const int bm = blockIdx.x * 128;
  const int bn = blockIdx.y * 128;

  v8f acc[2][4] = {};

  for (int k0 = 0; k0 < K; k0 += 16) {
    // Stage X tile 128x16 (float4 global loads, scalar LDS stores due to pad)
    for (int t = tid; t < 512; t += 256) {
      int r = t >> 2, c4 = t & 3;
      float4 v = *(const float4*)&X[(size_t)(bm + r) * K + k0 + c4 * 4];
      Xs[r][c4 * 4 + 0] = v.x;
      Xs[r][c4 * 4 + 1] = v.y;
      Xs[r][c4 * 4 + 2] = v.z;
      Xs[r][c4 * 4 + 3] = v.w;
    }
    // Stage W tile 16x128, transposed into LDS: Wst[col][k]
    for (int t = tid; t < 512; t += 256) {
      int r = t >> 5, c4 = t & 31;  // r = k row, c4 = col quad
      float4 v = *(const float4*)&Wt[(size_t)(k0 + r) * N + bn + c4 * 4];
      Wst[c4 * 4 + 0][r] = v.x;
      Wst[c4 * 4 + 1][r] = v.y;
      Wst[c4 * 4 + 2][r] = v.z;
      Wst[c4 * 4 + 3][r] = v.w;
    }
    __syncthreads();

#pragma unroll
    for (int kk = 0; kk < 16; kk += 4) {
      v2f a[2], bfr[4];
#pragma unroll
      for (int i = 0; i < 2; i++) {
        int row = wm * 32 + i * 16 + l16;
        a[i] = *(const v2f*)&Xs[row][kk + half2];      // 8B aligned
      }
#pragma unroll
      for (int j = 0; j < 4; j++) {
        int col = wn * 64 + j * 16 + l16;
        bfr[j] = *(const v2f*)&Wst[col][kk + half2];   // 8B aligned
      }
#pragma unroll
      for (int i = 0; i < 2; i++)
#pragma unroll
        for (int j = 0; j < 4; j++)
          acc[i][j] = wmma_f32_4(a[i], bfr[j], acc[i][j]);
    }
    __syncthreads();
  }

  // Epilogue: bias add + store. C layout: lanes 0-15 rows r, 16-31 rows r+8.
#pragma unroll
  for (int i = 0; i < 2; i++)
#pragma unroll
    for (int j = 0; j < 4; j++) {
      int col = bn + wn * 64 + j * 16 + l16;
      float bv = bias[col];
      int rowBase = bm + wm * 32 + i * 16 + ((lane >> 4) ? 8 : 0);
#pragma unroll
      for (int r = 0; r < 8; r++)
        Y[(size_t)(rowBase + r) * N + col] = acc[i][j][r] + bv;
    }
}

// ---------------------------------------------------------------------------
// Final GEMM: out = X(Mx256) * W3p(256x16) + b3p, store cols 0..7 only.
// Block tile 256x16, 8 waves stacked in M, wave tile 32x16 (2 subtiles).
// ---------------------------------------------------------------------------
__global__ __launch_bounds__(256) void gemm_out_kernel(
    const float* __restrict__ X, const float* __restrict__ Wt,
    const float* __restrict__ bias, float* __restrict__ out, int M, int K) {
  __shared__ float Xs[256][18];
  __shared__ float Wst[16][18];   // [col][k] transposed

  const int tid = threadIdx.x;
  const int wave = tid >> 5;
  const int lane = tid & 31;
  const int l16 = lane & 15;
  const int half2 = (lane >> 4) << 1;
  const int bm = blockIdx.x * 256;

  v8f acc[2] = {};

  for (int k0 = 0; k0 < K; k0 += 16) {
    for (int t = tid; t < 1024; t += 256) {
      int r = t >> 2, c4 = t & 3;
      float4 v = *(const float4*)&X[(size_t)(bm + r) * K + k0 + c4 * 4];
      Xs[r][c4 * 4 + 0] = v.x;
      Xs[r][c4 * 4 + 1] = v.y;
      Xs[r][c4 * 4 + 2] = v.z;
      Xs[r][c4 * 4 + 3] = v.w;
    }
    if (tid < 64) {
      int r = tid >> 2, c4 = tid & 3;  // r = k row, c4 = col quad
      float4 v = *(const float4*)&Wt[(size_t)(k0 + r) * 16 + c4 * 4];
      Wst[c4 * 4 + 0][r] = v.x;
      Wst[c4 * 4 + 1][r] = v.y;
      Wst[c4 * 4 + 2][r] = v.z;
      Wst[c4 * 4 + 3][r] = v.w;
    }
    __syncthreads();
#pragma unroll
    for (int kk = 0; kk < 16; kk += 4) {
      v2f bfr = *(const v2f*)&Wst[l16][kk + half2];
#pragma unroll
      for (int i = 0; i < 2; i++) {
        int row = wave * 32 + i * 16 + l16;
        v2f a = *(const v2f*)&Xs[row][kk + half2];
        acc[i] = wmma_f32_4(a, bfr, acc[i]);
      }
    }
    __syncthreads();
  }

  if (l16 < 8) {
    float bv = bias[l16];
#pragma unroll
    for (int i = 0; i < 2; i++) {
      int rowBase = bm + wave * 32 + i * 16 + ((lane >> 4) ? 8 : 0);
#pragma unroll
      for (int r = 0; r < 8; r++)
        out[(size_t)(rowBase + r) * 8 + l16] = acc[i][r] + bv;
    }
  }
}

// ---------------------------------------------------------------------------
// Deterministic column stats (two stage) + BN-ReLU
// ---------------------------------------------------------------------------
__global__ __launch_bounds__(256) void col_partial_kernel(
    const float* __restrict__ Y, float* __restrict__ psum,
    float* __restrict__ psq, int M, int N, int rowsPerBlock) {
  int col = threadIdx.x;
  int r0 = blockIdx.x * rowsPerBlock;
  int r1 = min(r0 + rowsPerBlock, M);
  float s = 0.f, sq = 0.f;
  for (int r = r0; r < r1; r++) {
    float v = Y[(size_t)r * N + col];
    s += v;
    sq += v * v;
  }
  psum[(size_t)blockIdx.x * N + col] = s;
  psq[(size_t)blockIdx.x * N + col] = sq;
}

__global__ __launch_bounds__(256) void col_finish_kernel(
    const float* __restrict__ psum, const float* __restrict__ psq,
    float* __restrict__ sum, float* __restrict__ sumsq, int nblk, int N) {
  int col = threadIdx.x;
  float s = 0.f, sq = 0.f;
  for (int b = 0; b < nblk; b++) {
    s += psum[(size_t)b * N + col];
    sq += psq[(size_t)b * N + col];
  }
  sum[col] = s;
  sumsq[col] = sq;
}

__global__ __launch_bounds__(256) void bn_relu_kernel(
    float* __restrict__ Y, const float* __restrict__ sum,
    const float* __restrict__ sumsq, const float* __restrict__ g,
    const float* __restrict__ be, int M, int N) {
  size_t total = (size_t)M * N;
  size_t stride = (size_t)gridDim.x * blockDim.x;
  for (size_t i = (size_t)blockIdx.x * blockDim.x + threadIdx.x; i < total;
       i += stride) {
    int col = (int)(i % (size_t)N);
    float mu = sum[col] / (float)M;
    float var = sumsq[col] / (float)M - mu * mu;
    float v = g[col] * (Y[i] - mu) * rsqrtf(var + 1e-5f) + be[col];
    Y[i] = v > 0.f ? v : 0.f;
  }
}

// ---------------------------------------------------------------------------
// Launcher
// ---------------------------------------------------------------------------
extern "C" void kernel_launch(void* const* d_in, const int* in_sizes, int n_in,
                              void* d_out, int out_size, void* d_ws,
                              size_t ws_size, hipStream_t stream) {
  const float* x   = (const float*)d_in[0];
  const float* W0  = (const float*)d_in[1];
  const float* b0  = (const float*)d_in[2];
  const float* g1  = (const float*)d_in[3];
  const float* be1 = (const float*)d_in[4];
  const float* W1  = (const float*)d_in[5];
  const float* b1  = (const float*)d_in[6];
  const float* g2  = (const float*)d_in[7];
  const float* be2 = (const float*)d_in[8];
  const float* W2  = (const float*)d_in[9];
  const float* b2  = (const float*)d_in[10];
  const float* g3  = (const float*)d_in[11];
  const float* be3 = (const float*)d_in[12];
  const float* W3  = (const float*)d_in[13];
  const float* b3  = (const float*)d_in[14];
  float* out = (float*)d_out;

  unsigned char* ws = (unsigned char*)d_ws;
  const size_t SZ_ACT = (size_t)MROWS * HID * sizeof(float);  // 64 MB
  float* bufA = (float*)(ws);
  float* bufB = (float*)(ws + SZ_ACT);
  size_t off = 2 * SZ_ACT;
  float* W0p = (float*)(ws + off); off += 256 * 256 * sizeof(float);
  float* W3p = (float*)(ws + off); off += 256 * 16 * sizeof(float);
  float* b3p = (float*)(ws + off); off += 64;
  float* sum = (float*)(ws + off); off += 1024;
  float* ssq = (float*)(ws + off); off += 1024;
  float* psum = (float*)(ws + off); off += 256 * 256 * sizeof(float);
  float* psq  = (float*)(ws + off); off += 256 * 256 * sizeof(float);

  const int RPB = MROWS / 256;  // rows per stats block

  // 0) pad weights
  prep_kernel<<<64, 256, 0, stream>>>(W0, W3, b3, W0p, W3p, b3p);

  // 1) geometry / spectral features -> bufA (M x 256, padded)
  geom_kernel<<<MROWS, 32, 0, stream>>>(x, bufA);

  dim3 gg(MROWS / 128, HID / 128);

  // 2) layer 1: bufA -> bufB
  gemm_wmma_kernel<<<gg, 256, 0, stream>>>(bufA, W0p, b0, bufB, MROWS, HID, HID);
  col_partial_kernel<<<256, 256, 0, stream>>>(bufB, psum, psq, MROWS, HID, RPB);
  col_finish_kernel<<<1, 256, 0, stream>>>(psum, psq, sum, ssq, 256, HID);
  bn_relu_kernel<<<2048, 256, 0, stream>>>(bufB, sum, ssq, g1, be1, MROWS, HID);

  // 3) layer 2: bufB -> bufA
  gemm_wmma_kernel<<<gg, 256, 0, stream>>>(bufB, W1, b1, bufA, MROWS, HID, HID);
  col_partial_kernel<<<256, 256, 0, stream>>>(bufA, psum, psq, MROWS, HID, RPB);
  col_finish_kernel<<<1, 256, 0, stream>>>(psum, psq, sum, ssq, 256, HID);
  bn_relu_kernel<<<2048, 256, 0, stream>>>(bufA, sum, ssq, g2, be2, MROWS, HID);

  // 4) layer 3: bufA -> bufB
  gemm_wmma_kernel<<<gg, 256, 0, stream>>>(bufA, W2, b2, bufB, MROWS, HID, HID);
  col_partial_kernel<<<256, 256, 0, stream>>>(bufB, psum, psq, MROWS, HID, RPB);
  col_finish_kernel<<<1, 256, 0, stream>>>(psum, psq, sum, ssq, 256, HID);
  bn_relu_kernel<<<2048, 256, 0, stream>>>(bufB, sum, ssq, g3, be3, MROWS, HID);

  // 5) output layer: bufB -> out (M x 8)
  gemm_out_kernel<<<MROWS / 256, 256, 0, stream>>>(bufB, W3p, b3p, out, MROWS,
                                                   HID);
}